// ImprovedGAT_84885733638250
// MI455X (gfx1250) — compile-verified
//
#include <hip/hip_runtime.h>
#include <hip/hip_bf16.h>

typedef __attribute__((ext_vector_type(2))) float v2f;
typedef __attribute__((ext_vector_type(8))) float v8f;

#define N_NODES 10000
#define N_EDGES 320000
#define E_TOT   (N_EDGES + N_NODES)   // self-loops appended
#define GS_BLOCKS 4096

// ---------------------------------------------------------------------------
// fp32 GEMM via V_WMMA_F32_16X16X4_F32. One wave computes a 16 x (16*NT) strip,
// reusing a single A fragment across NT accumulators per k-step.
// A: [M,K] row-major, B: [K,N] row-major, D: [M,N]. M%16==0, N%(16*NT)==0, K%4==0.
// A frag (16x4): lanes 0-15 hold M=lane,K={k0,k0+1}; lanes 16-31 M=lane-16,K={k0+2,k0+3}
// B frag (4x16): lanes 0-15 hold N=lane,K={k0,k0+1}; lanes 16-31 N=lane-16,K={k0+2,k0+3}
// D: VGPR v, lanes 0-15 -> (M=m0+v, N); lanes 16-31 -> (M=m0+8+v, N)
// ---------------------------------------------------------------------------
template <int NT>
__global__ void gemm_wmma_f32(const float* __restrict__ A, const float* __restrict__ B,
                              float* __restrict__ D, int N, int K) {
    const int m0   = blockIdx.x * 16;
    const int n0   = blockIdx.y * (16 * NT);
    const int lane = threadIdx.x;        // 0..31 (wave32)
    const int half = lane >> 4;          // 0 or 1
    const int l15  = lane & 15;

    v8f acc[NT];
#pragma unroll
    for (int j = 0; j < NT; ++j) acc[j] = (v8f){};

    const float* __restrict__ arow = A + (long)(m0 + l15) * K;
    const float* __restrict__ bcol = B + (n0 + l15);
    for (int k0 = 0; k0 < K; k0 += 4) {
        const int kk = k0 + half * 2;
        v2f a; a.x = arow[kk]; a.y = arow[kk + 1];
        const float* __restrict__ brow0 = bcol + (long)kk * N;
        const float* __restrict__ brow1 = bcol + (long)(kk + 1) * N;
#pragma unroll
        for (int j = 0; j < NT; ++j) {
            v2f b; b.x = brow0[16 * j]; b.y = brow1[16 * j];
            acc[j] = __builtin_amdgcn_wmma_f32_16x16x4_f32(
                         /*neg_a=*/false, a, /*neg_b=*/false, b,
                         /*c_mod=*/(short)0, acc[j], /*reuse_a=*/false, /*reuse_b=*/false);
        }
    }
    const int rbase = m0 + half * 8;
#pragma unroll
    for (int j = 0; j < NT; ++j) {
        const int col = n0 + 16 * j + l15;
#pragma unroll
        for (int v = 0; v < 8; ++v)
            D[(long)(rbase + v) * N + col] = acc[j][v];
    }
}

// ---------------------------------------------------------------------------
__global__ void fill_f32(float* __restrict__ p, float v, long n) {
    long i = (long)blockIdx.x * blockDim.x + threadIdx.x;
    if (i < n) p[i] = v;
}

__device__ inline void atomicMaxF(float* addr, float v) {
    // IEEE trick: positives via signed-int max, negatives via unsigned-int min
    if (v >= 0.0f) atomicMax((int*)addr, __float_as_int(v));
    else           atomicMin((unsigned int*)addr, __float_as_uint(v));
}

__device__ inline void edge_ends(const int* __restrict__ ei, long e, int& s, int& d) {
    if (e < N_EDGES) { s = ei[e]; d = ei[N_EDGES + e]; }
    else             { s = d = (int)(e - N_EDGES); }
}

// ---------------------------------------------------------------------------
// Per-edge attention scores: one wave per edge. H heads of 32 channels each.
// Lane covers H contiguous channels starting at lane*H (all in head lane*H/32).
// score[e,h] = sum_c att[h*32+c] * leaky_relu(xl[src,·]+xr[dst,·], 0.2)
// Also atomic running max into m[dst,h].
// ---------------------------------------------------------------------------
template <int H>
__global__ void edge_scores_k(const float* __restrict__ xl, const float* __restrict__ xr,
                              const float* __restrict__ att, const int* __restrict__ ei,
                              float* __restrict__ score, float* __restrict__ m) {
    constexpr int HC = H * 32;
    const int  lane = threadIdx.x & 31;
    const long wid  = ((long)blockIdx.x * blockDim.x + threadIdx.x) >> 5;
    const long nw   = ((long)gridDim.x * blockDim.x) >> 5;
    for (long e = wid; e < E_TOT; e += nw) {
        int s, d; edge_ends(ei, e, s, d);
        const float* __restrict__ pl = xl + (long)s * HC;
        const float* __restrict__ pr = xr + (long)d * HC;
        float sum = 0.0f;
#pragma unroll
        for (int j = 0; j < H; ++j) {
            const int ch = lane * H + j;
            float z = pl[ch] + pr[ch];
            z = (z > 0.0f) ? z : 0.2f * z;
            sum += att[ch] * z;
        }
        constexpr int group = 32 / H;   // lanes per head
#pragma unroll
        for (int off = 1; off < group; off <<= 1)
            sum += __shfl_xor(sum, off, 32);
        if ((lane & (group - 1)) == 0) {
            const int h = lane / group;
            score[e * H + h] = sum;
            atomicMaxF(&m[(long)d * H + h], sum);
        }
    }
}

// exp(score - max) in place; atomic denom accumulation
template <int H>
__global__ void edge_softmax_k(float* __restrict__ score, const float* __restrict__ m,
                               float* __restrict__ denom, const int* __restrict__ ei) {
    const long tot = (long)E_TOT * H;
    for (long i = (long)blockIdx.x * blockDim.x + threadIdx.x; i < tot;
         i += (long)gridDim.x * blockDim.x) {
        const long e = i / H;
        const int  h = (int)(i - e * H);
        const int  d = (e < N_EDGES) ? ei[N_EDGES + e] : (int)(e - N_EDGES);
        const float ex = __expf(score[i] - m[(long)d * H + h]);
        score[i] = ex;
        atomicAdd(&denom[(long)d * H + h], ex);
    }
}

// score[e,h] <- score[e,h] / (denom[dst,h] + eps)  (finalize alpha once per edge-head)
template <int H>
__global__ void alpha_k(float* __restrict__ score, const float* __restrict__ denom,
                        const int* __restrict__ ei) {
    const long tot = (long)E_TOT * H;
    for (long i = (long)blockIdx.x * blockDim.x + threadIdx.x; i < tot;
         i += (long)gridDim.x * blockDim.x) {
        const long e = i / H;
        const int  h = (int)(i - e * H);
        const int  d = (e < N_EDGES) ? ei[N_EDGES + e] : (int)(e - N_EDGES);
        score[i] = score[i] / (denom[(long)d * H + h] + 1e-16f);
    }
}

// out[dst, ch] += alpha[e,h] * xl[src, ch]  — flat (edge, channel) threads
template <int H>
__global__ void aggregate_k(const float* __restrict__ alpha, const float* __restrict__ xl,
                            const int* __restrict__ ei, float* __restrict__ out) {
    constexpr int HC = H * 32;
    const long tot = (long)E_TOT * HC;
    for (long i = (long)blockIdx.x * blockDim.x + threadIdx.x; i < tot;
         i += (long)gridDim.x * blockDim.x) {
        const long e  = i / HC;
        const int  ch = (int)(i - e * HC);
        const int  h  = ch >> 5;
        int s, d; edge_ends(ei, e, s, d);
        atomicAdd(&out[(long)d * HC + ch], alpha[e * H + h] * xl[(long)s * HC + ch]);
    }
}

// mode 0: out = relu(agg + b); mode 1: out = agg + b + res; mode 2: out = agg + b
__global__ void epilogue_k(const float* __restrict__ agg, const float* __restrict__ bias,
                           const float* __restrict__ res, float* __restrict__ out,
                           long n, int F, int mode) {
    long i = (long)blockIdx.x * blockDim.x + threadIdx.x;
    if (i >= n) return;
    float v = agg[i] + bias[i % F];
    if (mode == 0)      v = (v > 0.0f) ? v : 0.0f;
    else if (mode == 1) v += res[i];
    out[i] = v;
}

// ---------------------------------------------------------------------------
static void gat_layer(const float* xin, int K, int H,
                      const float* Wl, const float* Wr, const float* att,
                      const float* bias, const float* res, float* out, int mode,
                      float* xl, float* xr, float* score, float* m, float* denom,
                      float* agg, const int* ei, hipStream_t stream) {
    const int  HC   = H * 32;
    const long nmh  = (long)N_NODES * H;
    const long nagg = (long)N_NODES * HC;

    if (H == 8) {
        dim3 gg(N_NODES / 16, HC / 64);
        gemm_wmma_f32<4><<<gg, 32, 0, stream>>>(xin, Wl, xl, HC, K);
        gemm_wmma_f32<4><<<gg, 32, 0, stream>>>(xin, Wr, xr, HC, K);
    } else {
        dim3 gg(N_NODES / 16, 1);
        gemm_wmma_f32<2><<<gg, 32, 0, stream>>>(xin, Wl, xl, HC, K);
        gemm_wmma_f32<2><<<gg, 32, 0, stream>>>(xin, Wr, xr, HC, K);
    }

    fill_f32<<<(int)((nmh  + 255) / 256), 256, 0, stream>>>(m,     -INFINITY, nmh);
    fill_f32<<<(int)((nmh  + 255) / 256), 256, 0, stream>>>(denom, 0.0f,      nmh);
    fill_f32<<<(int)((nagg + 255) / 256), 256, 0, stream>>>(agg,   0.0f,      nagg);

    if (H == 8) {
        edge_scores_k<8> <<<GS_BLOCKS, 256, 0, stream>>>(xl, xr, att, ei, score, m);
        edge_softmax_k<8><<<GS_BLOCKS, 256, 0, stream>>>(score, m, denom, ei);
        alpha_k<8>       <<<GS_BLOCKS, 256, 0, stream>>>(score, denom, ei);
        aggregate_k<8>   <<<GS_BLOCKS, 256, 0, stream>>>(score, xl, ei, agg);
    } else {
        edge_scores_k<1> <<<GS_BLOCKS, 256, 0, stream>>>(xl, xr, att, ei, score, m);
        edge_softmax_k<1><<<GS_BLOCKS, 256, 0, stream>>>(score, m, denom, ei);
        alpha_k<1>       <<<GS_BLOCKS, 256, 0, stream>>>(score, denom, ei);
        aggregate_k<1>   <<<GS_BLOCKS, 256, 0, stream>>>(score, xl, ei, agg);
    }

    epilogue_k<<<(int)((nagg + 255) / 256), 256, 0, stream>>>(agg, bias, res, out,
                                                              nagg, HC, mode);
}

extern "C" void kernel_launch(void* const* d_in, const int* in_sizes, int n_in,
                              void* d_out, int out_size, void* d_ws, size_t ws_size,
                              hipStream_t stream) {
    const float* x   = (const float*)d_in[0];
    const float* W1l = (const float*)d_in[1];
    const float* W1r = (const float*)d_in[2];
    const float* a1  = (const float*)d_in[3];
    const float* b1  = (const float*)d_in[4];
    const float* W2l = (const float*)d_in[5];
    const float* W2r = (const float*)d_in[6];
    const float* a2  = (const float*)d_in[7];
    const float* b2  = (const float*)d_in[8];
    const float* W3l = (const float*)d_in[9];
    const float* W3r = (const float*)d_in[10];
    const float* a3  = (const float*)d_in[11];
    const float* b3  = (const float*)d_in[12];
    const int*   ei  = (const int*)d_in[13];   // [2, E] int32 (JAX x64 off)

    constexpr int HID = 256, H = 8;
    float* ws    = (float*)d_ws;
    float* xl    = ws; ws += (size_t)N_NODES * HID;
    float* xr    = ws; ws += (size_t)N_NODES * HID;
    float* agg   = ws; ws += (size_t)N_NODES * HID;
    float* h1    = ws; ws += (size_t)N_NODES * HID;
    float* h2    = ws; ws += (size_t)N_NODES * HID;
    float* score = ws; ws += (size_t)E_TOT * H;
    float* m     = ws; ws += (size_t)N_NODES * H;
    float* denom = ws; ws += (size_t)N_NODES * H;

    // Layer 1: 128 -> 8x32, ReLU
    gat_layer(x, 128, 8, W1l, W1r, a1, b1, nullptr, h1, /*relu*/0,
              xl, xr, score, m, denom, agg, ei, stream);
    // Layer 2: 256 -> 8x32, + residual h1
    gat_layer(h1, HID, 8, W2l, W2r, a2, b2, h1, h2, /*res*/1,
              xl, xr, score, m, denom, agg, ei, stream);
    // Layer 3: 256 -> 1x32, plain bias, into d_out
    gat_layer(h2, HID, 1, W3l, W3r, a3, b3, nullptr, (float*)d_out, /*plain*/2,
              xl, xr, score, m, denom, agg, ei, stream);

    (void)in_sizes; (void)n_in; (void)out_size; (void)ws_size;
}